// BERT_CRF_fusion_emb_10316511445786
// MI455X (gfx1250) — compile-verified
//
#include <hip/hip_runtime.h>
#include <hip/hip_bf16.h>
#include <math.h>

// Problem dims (fixed by the reference)
#define B_  64
#define S_  512
#define H_  768
#define T_  9

typedef float v2f __attribute__((ext_vector_type(2)));
typedef float v8f __attribute__((ext_vector_type(8)));

__device__ __forceinline__ float wave_sum32(float v) {
  #pragma unroll
  for (int o = 16; o > 0; o >>= 1) v += __shfl_xor(v, o, 32);
  return v;
}
__device__ __forceinline__ float wave_max32(float v) {
  #pragma unroll
  for (int o = 16; o > 0; o >>= 1) v = fmaxf(v, __shfl_xor(v, o, 32));
  return v;
}
__device__ __forceinline__ int wave_isum32(int v) {
  #pragma unroll
  for (int o = 16; o > 0; o >>= 1) v += __shfl_xor(v, o, 32);
  return v;
}

// ---------------------------------------------------------------------------
// Kernel A: ext-emb gather + gate + fuse + (fused @ cls_w + cls_b)
// One wave handles a 16-token tile; the 768->9 GEMM runs on
// V_WMMA_F32_16X16X4_F32 (f32 in, f32 acc: no precision loss; we are
// bandwidth-bound so f32 matrix throughput is more than enough).
// ---------------------------------------------------------------------------
__global__ __launch_bounds__(256) void fused_logits_kernel(
    const float* __restrict__ seq,     // [B*S, H]
    const float* __restrict__ table,   // [V_EXT, H]
    const float* __restrict__ gate_w,  // [2H]
    const float* __restrict__ gate_b,  // [1]
    const float* __restrict__ cls_w,   // [H, T]
    const float* __restrict__ cls_b,   // [T]
    const int*   __restrict__ ids,     // [B*S]
    float* __restrict__ logits)        // [B*S, T]
{
  // cls_w staged in LDS, padded 9 -> 16 columns with zeros (no branch, no
  // bank conflicts: lanes 0-15 and 16-31 hit disjoint bank groups).
  __shared__ float s_clsw[H_ * 16];
  __shared__ float s_gw[2 * H_];

  const int tid = threadIdx.x;
  for (int i = tid; i < H_ * 16; i += 256) {
    int r = i >> 4, c = i & 15;
    s_clsw[i] = (c < T_) ? cls_w[r * T_ + c] : 0.0f;
  }
  for (int i = tid; i < 2 * H_; i += 256) s_gw[i] = gate_w[i];
  __syncthreads();

  const int lane = tid & 31;
  const int wave = tid >> 5;
  const int tile = blockIdx.x * 8 + wave;  // 16 rows per tile
  const int base = tile * 16;
  const float gb = gate_b[0];

  // --- per-row gate, wave-cooperative 1536-wide dot + butterfly reduce ---
  const int myrow = base + (lane & 15);
  const int myidx = ids[myrow];
  float my_g = 0.0f;
  for (int m = 0; m < 16; ++m) {
    const int row = base + m;
    const float* sp = seq + (size_t)row * H_;
    const float* ep = table + (size_t)ids[row] * H_;
    float acc = 0.0f;
    for (int k = lane; k < H_; k += 32)
      acc = fmaf(sp[k], s_gw[k], fmaf(ep[k], s_gw[H_ + k], acc));
    acc = wave_sum32(acc);                       // all lanes hold the sum
    float g = 1.0f / (1.0f + __expf(-(acc + gb)));
    if ((lane & 15) == m) my_g = g;              // keep only my row's gate
  }

  // --- WMMA f32 16x16x4 over K = H ---
  // A layout: lane L<16 holds A[M=L][k+0],A[M=L][k+1]; lane L+16 holds k+2,k+3
  // B layout: lane L<16 holds B[k+0][n],B[k+1][n]; lane L+16 holds k+2,k+3
  const int   n    = lane & 15;
  const int   koff = (lane >> 4) * 2;
  const float* sp  = seq + (size_t)myrow * H_ + koff;
  const float* ep  = table + (size_t)myidx * H_ + koff;
  const float omg  = 1.0f - my_g;

  v8f c;
  #pragma unroll
  for (int v = 0; v < 8; ++v) c[v] = 0.0f;

  for (int kk = 0; kk < H_; kk += 4) {
    float2 sv = *reinterpret_cast<const float2*>(sp + kk);  // 8B aligned
    float2 ev = *reinterpret_cast<const float2*>(ep + kk);
    v2f a, b;
    a[0] = my_g * sv.x + omg * ev.x;   // fused = g*seq + (1-g)*ext
    a[1] = my_g * sv.y + omg * ev.y;
    b[0] = s_clsw[(kk + koff) * 16 + n];
    b[1] = s_clsw[(kk + koff + 1) * 16 + n];
    // (neg_a, A, neg_b, B, c_mod, C, reuse_a, reuse_b)
    c = __builtin_amdgcn_wmma_f32_16x16x4_f32(false, a, false, b,
                                              (short)0, c, false, false);
  }

  // C/D layout: lanes 0-15 -> M = v, lanes 16-31 -> M = v+8, N = lane&15
  if (n < T_) {
    const float bias = cls_b[n];
    #pragma unroll
    for (int v = 0; v < 8; ++v) {
      int m = (lane < 16) ? v : (v + 8);
      logits[(size_t)(base + m) * T_ + n] = c[v] + bias;
    }
  }
}

// ---------------------------------------------------------------------------
// Kernel B: CRF negative log-likelihood. One wave per batch element.
// The mask is a prefix (arange(S) < length), so the reference's stable sort
// is the identity and the scan simply stops at `len`.
// Lanes 0..8 own the 9 alpha states; alphas are exchanged via __shfl.
// ---------------------------------------------------------------------------
__global__ __launch_bounds__(32) void crf_nll_kernel(
    const float* __restrict__ logits,  // [B, S, T]
    const int*   __restrict__ labels,  // [B, S]
    const float* __restrict__ start_trans,  // [T]
    const float* __restrict__ end_trans,    // [T]
    const float* __restrict__ trans,        // [T, T]
    float* __restrict__ out)                // [1]
{
  const int b = blockIdx.x;
  const int j = threadIdx.x;  // lane
  const int*   lab = labels + (size_t)b * S_;
  const float* lg  = logits + (size_t)b * S_ * T_;

  // sequence length = number of valid labels (prefix mask)
  int cnt = 0;
  for (int t = j; t < S_; t += 32) cnt += (lab[t] != -100) ? 1 : 0;
  const int len = wave_isum32(cnt);   // >= S/2 >= 1

  // ----- numerator -----
  float s1 = 0.0f, s2 = 0.0f;
  for (int t = j; t < len; t += 32) {
    const int tg = lab[t];
    s1 += lg[(size_t)t * T_ + tg];
    if (t >= 1) s2 += trans[lab[t - 1] * T_ + tg];
  }
  float num = wave_sum32(s1) + wave_sum32(s2);
  num += start_trans[lab[0]] + end_trans[lab[len - 1]];

  // ----- denominator: forward algorithm -----
  float trcol[T_];
  float alpha = -1e30f;
  if (j < T_) {
    #pragma unroll
    for (int i = 0; i < T_; ++i) trcol[i] = trans[i * T_ + j];
    alpha = start_trans[j] + lg[j];
  } else {
    #pragma unroll
    for (int i = 0; i < T_; ++i) trcol[i] = 0.0f;
  }

  for (int t = 1; t < len; ++t) {
    float av[T_];
    #pragma unroll
    for (int i = 0; i < T_; ++i) av[i] = __shfl(alpha, i, 32); // uniform exec
    if (j < T_) {
      float m = -1e30f;
      #pragma unroll
      for (int i = 0; i < T_; ++i) m = fmaxf(m, av[i] + trcol[i]);
      float s = 0.0f;
      #pragma unroll
      for (int i = 0; i < T_; ++i) s += __expf(av[i] + trcol[i] - m);
      alpha = m + __logf(s) + lg[(size_t)t * T_ + j];
    }
  }

  const float val = (j < T_) ? (alpha + end_trans[j]) : -1e30f;
  const float mx = wave_max32(val);
  const float se = wave_sum32((j < T_) ? __expf(val - mx) : 0.0f);
  const float denom = mx + __logf(se);

  if (j == 0) atomicAdd(out, (denom - num) * (1.0f / (float)B_));
}

// ---------------------------------------------------------------------------
extern "C" void kernel_launch(void* const* d_in, const int* in_sizes, int n_in,
                              void* d_out, int out_size, void* d_ws, size_t ws_size,
                              hipStream_t stream) {
  (void)in_sizes; (void)n_in; (void)out_size; (void)ws_size;
  const float* seq   = (const float*)d_in[0];   // sequence_output
  const float* table = (const float*)d_in[1];   // ext_emb_table
  const float* gw    = (const float*)d_in[2];   // gate_w
  const float* gb    = (const float*)d_in[3];   // gate_b
  const float* cw    = (const float*)d_in[4];   // cls_w
  const float* cb    = (const float*)d_in[5];   // cls_b
  const float* st    = (const float*)d_in[6];   // start_trans
  const float* et    = (const float*)d_in[7];   // end_trans
  const float* tr    = (const float*)d_in[8];   // trans
  const int*   ids   = (const int*)d_in[9];     // extended_input_ids
  const int*   lab   = (const int*)d_in[10];    // labels

  float* logits = (float*)d_ws;                 // [B*S, T] = 1.18 MB scratch

  hipMemsetAsync(d_out, 0, sizeof(float), stream);

  // 2048 tiles of 16 tokens, 8 waves (256 threads) per block -> 256 blocks
  fused_logits_kernel<<<dim3((B_ * S_ / 16) / 8), 256, 0, stream>>>(
      seq, table, gw, gb, cw, cb, ids, logits);

  crf_nll_kernel<<<dim3(B_), 32, 0, stream>>>(
      logits, lab, st, et, tr, (float*)d_out);
}